// Decoder_82214263980175
// MI455X (gfx1250) — compile-verified
//
#include <hip/hip_runtime.h>
#include <math.h>

#define Nn 32
#define Ss 128
#define Tt 48
#define Ee 256
#define Hh 512
#define Vv 32000
#define Gg 2048            // 4*H
#define KX 1280            // (E + H) + H
#define ROWS (Nn*Tt)       // 1536
#define CHUNK 256
#define NCH (Vv/CHUNK)     // 125
#define KBLK 32
#define ROWT 32
#define PITCH2 17          // pitch in v2f units (= 34 floats, even => 8B-aligned pairs)
#define LPITCH 257         // logits tile pitch (floats)

typedef float v2f __attribute__((ext_vector_type(2)));
typedef float v8f __attribute__((ext_vector_type(8)));

__device__ __forceinline__ float sigf(float x) { return 1.0f / (1.0f + __expf(-x)); }

// ---------------- attention: one block per batch element ----------------
__global__ __launch_bounds__(128) void attn_kernel(const float* __restrict__ enc,
                                                   const int* __restrict__ src_len,
                                                   const float* __restrict__ h,
                                                   float* __restrict__ context) {
    __shared__ float sh_h[Hh];
    __shared__ float attn[Ss];
    __shared__ float red[Ss];
    int n = blockIdx.x;
    int s = threadIdx.x;           // 128 threads == S
    for (int i = s; i < Hh; i += Ss) sh_h[i] = h[n * Hh + i];
    __syncthreads();
    const float* e = enc + ((size_t)(n * Ss + s)) * Hh;
    float acc = 0.0f;
    for (int k = 0; k < Hh; ++k) acc += sh_h[k] * e[k];
    acc *= 0.044194173824159216f;  // 1/sqrt(512)
    if (s >= src_len[n]) acc = -10000000.0f;
    red[s] = acc;
    __syncthreads();
    for (int off = 64; off > 0; off >>= 1) {
        if (s < off) red[s] = fmaxf(red[s], red[s + off]);
        __syncthreads();
    }
    float m = red[0];
    __syncthreads();
    float ex = __expf(acc - m);
    red[s] = ex;
    __syncthreads();
    for (int off = 64; off > 0; off >>= 1) {
        if (s < off) red[s] += red[s + off];
        __syncthreads();
    }
    float inv = 1.0f / red[0];
    attn[s] = ex * inv;
    __syncthreads();
    // context: each thread owns 4 h-values
    for (int j = 0; j < 4; ++j) {
        int hh = s + j * Ss;
        float a = 0.0f;
        for (int ss = 0; ss < Ss; ++ss)
            a += attn[ss] * enc[((size_t)(n * Ss + ss)) * Hh + hh];
        context[n * Hh + hh] = a;
    }
}

// ---------------- pack x = [emb[target[:,t]], context, h] -> xbuf (N x 1280)
__global__ __launch_bounds__(256) void xpack_kernel(const float* __restrict__ emb,
                                                    const int* __restrict__ target,
                                                    const float* __restrict__ context,
                                                    const float* __restrict__ h,
                                                    float* __restrict__ xbuf, int t) {
    int idx = blockIdx.x * 256 + threadIdx.x;
    if (idx >= Nn * KX) return;
    int m = idx / KX, k = idx % KX;
    float v;
    if (k < Ee)           v = emb[(size_t)target[m * Tt + t] * Ee + k];
    else if (k < Ee + Hh) v = context[m * Hh + (k - Ee)];
    else                  v = h[m * Hh + (k - Ee - Hh)];
    xbuf[idx] = v;
}

// ---------------- gates = xbuf @ [W_ih|W_hh]^T  via fp32 WMMA --------------
// grid 32 blocks x 256 threads; one 16x16 tile per wave (2 mtiles x 128 ctiles)
__global__ __launch_bounds__(256) void gates_gemm_kernel(const float* __restrict__ xbuf,
                                                         const float* __restrict__ W_ih,
                                                         const float* __restrict__ W_hh,
                                                         float* __restrict__ gates) {
    int wave = blockIdx.x * 8 + (threadIdx.x >> 5);  // 0..255
    int lane = threadIdx.x & 31;
    int mtile = wave >> 7;     // 0..1
    int ctile = wave & 127;    // 0..127
    int nloc  = lane & 15;
    int khalf = (lane >> 4) * 2;
    int m   = mtile * 16 + nloc;
    int col = ctile * 16 + nloc;
    const float* xrow = xbuf + (size_t)m * KX;
    const float* wih  = W_ih + (size_t)col * 768;
    const float* whh  = W_hh + (size_t)col * Hh;
    v8f acc; for (int i = 0; i < 8; ++i) acc[i] = 0.0f;
    for (int kb = 0; kb < KX; kb += 4) {
        int k0 = kb + khalf;
        v2f a; a.x = xrow[k0]; a.y = xrow[k0 + 1];
        v2f b;
        if (kb < 768) { b.x = wih[k0];       b.y = wih[k0 + 1]; }
        else          { b.x = whh[k0 - 768]; b.y = whh[k0 - 767]; }
        acc = __builtin_amdgcn_wmma_f32_16x16x4_f32(false, a, false, b,
                                                    (short)0, acc, false, false);
    }
    int mbase = mtile * 16 + ((lane >> 4) ? 8 : 0);
    for (int v = 0; v < 8; ++v)
        gates[(size_t)(mbase + v) * Gg + ctile * 16 + nloc] = acc[v];
}

// ---------------- LSTM cell update -------------------------------------
__global__ __launch_bounds__(256) void lstm_update_kernel(const float* __restrict__ gates,
                                                          const float* __restrict__ b_ih,
                                                          const float* __restrict__ b_hh,
                                                          float* __restrict__ h,
                                                          float* __restrict__ c,
                                                          float* __restrict__ dec_out, int t) {
    int idx = blockIdx.x * 256 + threadIdx.x;   // N*H = 16384
    if (idx >= Nn * Hh) return;
    int n = idx / Hh, hh = idx % Hh;
    const float* g = gates + (size_t)n * Gg;
    float ig = g[hh]            + b_ih[hh]            + b_hh[hh];
    float fg = g[Hh + hh]       + b_ih[Hh + hh]       + b_hh[Hh + hh];
    float gg = g[2 * Hh + hh]   + b_ih[2 * Hh + hh]   + b_hh[2 * Hh + hh];
    float og = g[3 * Hh + hh]   + b_ih[3 * Hh + hh]   + b_hh[3 * Hh + hh];
    float cn = sigf(fg) * c[idx] + sigf(ig) * tanhf(gg);
    float hn = sigf(og) * tanhf(cn);
    c[idx] = cn;
    h[idx] = hn;
    dec_out[((size_t)(n * Tt + t)) * Hh + hh] = hn;
}

// ---------------- fused logits GEMM + online softmax partials ----------
// grid (NCH=125, ROWS/32=48); 256 threads = 8 waves.
// Block tile: 32 rows x 256 cols. Each wave: 2x2 grid of 16x16 WMMA tiles
// (rows {0-15,16-31}, cols {w*32..+15, +16..+31}) -> A/B frags each reused
// twice; frags are single aligned ds_load_b64 thanks to even-pitch layouts.
__global__ __launch_bounds__(256) void logits_kernel(const float* __restrict__ dec_out,
                                                     const float* __restrict__ W_out,
                                                     const float* __restrict__ b_out,
                                                     const int* __restrict__ target,
                                                     float* __restrict__ pmax,
                                                     float* __restrict__ psum,
                                                     int* __restrict__ pidx,
                                                     float* __restrict__ ptgt) {
    __shared__ v2f A2[ROWT * PITCH2];     // [m][k/2], 34-float pitch  (4.3 KB)
    __shared__ v2f B2[CHUNK * PITCH2];    // [col][k/2]                (34.8 KB)
    __shared__ float smax[256];
    __shared__ int   sidx[256];
    float* A_lds = (float*)A2;
    float* B_lds = (float*)B2;

    int ch = blockIdx.x, rt = blockIdx.y;
    int rowbase = rt * ROWT, colbase = ch * CHUNK;
    int tid = threadIdx.x;
    int lane = tid & 31, wave = tid >> 5;
    int nloc = lane & 15;
    int kup  = lane >> 4;                 // upper half-lanes take k+2
    int wcol0 = wave * 32 + nloc;
    int wcol1 = wcol0 + 16;
    v8f acc00, acc01, acc10, acc11;
    for (int i = 0; i < 8; ++i) { acc00[i] = 0.0f; acc01[i] = 0.0f;
                                  acc10[i] = 0.0f; acc11[i] = 0.0f; }

    for (int kb = 0; kb < Hh; kb += KBLK) {
        // stage A: 32 rows x 32 k (coalesced global, stride-1 LDS writes)
        for (int i = tid; i < ROWT * KBLK; i += 256) {
            int m = i >> 5, k = i & (KBLK - 1);
            A_lds[m * (2 * PITCH2) + k] = dec_out[(size_t)(rowbase + m) * Hh + kb + k];
        }
        // stage B: 256 cols x 32 k
        for (int i = tid; i < CHUNK * KBLK; i += 256) {
            int cc = i >> 5, kk = i & (KBLK - 1);
            B_lds[cc * (2 * PITCH2) + kk] = W_out[(size_t)(colbase + cc) * Hh + kb + kk];
        }
        __syncthreads();
        for (int k4 = 0; k4 < KBLK; k4 += 4) {
            int kh = (k4 >> 1) + kup;     // index in v2f units: k0 = k4 + 2*kup
            v2f a0 = A2[nloc * PITCH2 + kh];
            v2f a1 = A2[(16 + nloc) * PITCH2 + kh];
            v2f b0 = B2[wcol0 * PITCH2 + kh];
            v2f b1 = B2[wcol1 * PITCH2 + kh];
            acc00 = __builtin_amdgcn_wmma_f32_16x16x4_f32(false, a0, false, b0,
                                                          (short)0, acc00, false, false);
            acc01 = __builtin_amdgcn_wmma_f32_16x16x4_f32(false, a0, false, b1,
                                                          (short)0, acc01, false, false);
            acc10 = __builtin_amdgcn_wmma_f32_16x16x4_f32(false, a1, false, b0,
                                                          (short)0, acc10, false, false);
            acc11 = __builtin_amdgcn_wmma_f32_16x16x4_f32(false, a1, false, b1,
                                                          (short)0, acc11, false, false);
        }
        __syncthreads();   // protect LDS reuse next iteration
    }
    // dump logits tile (+bias) into B_lds reused as L[m][col], pitch LPITCH
    float* L = B_lds;                    // 32*257 = 8224 floats <= 8704 available
    int mb = kup ? 8 : 0;
    float bo0 = b_out[colbase + wcol0];
    float bo1 = b_out[colbase + wcol1];
    for (int v = 0; v < 8; ++v) {
        L[(mb + v) * LPITCH + wcol0]        = acc00[v] + bo0;
        L[(mb + v) * LPITCH + wcol1]        = acc01[v] + bo1;
        L[(16 + mb + v) * LPITCH + wcol0]   = acc10[v] + bo0;
        L[(16 + mb + v) * LPITCH + wcol1]   = acc11[v] + bo1;
    }
    __syncthreads();
    // per-row chunk stats: 8 threads per row (32 rows)
    int r = tid >> 3, j = tid & 7;
    int row = rowbase + r;
    float lmax = -1e30f; int lidx = 0x7fffffff;
    for (int ci = j; ci < CHUNK; ci += 8) {
        float vl = L[r * LPITCH + ci];
        int gi = colbase + ci;
        if (vl > lmax || (vl == lmax && gi < lidx)) { lmax = vl; lidx = gi; }
    }
    smax[tid] = lmax; sidx[tid] = lidx;
    __syncthreads();
    if (j == 0) {
        float bm = smax[tid]; int bi = sidx[tid];
        for (int q = 1; q < 8; ++q) {
            float vq = smax[tid + q]; int iq = sidx[tid + q];
            if (vq > bm || (vq == bm && iq < bi)) { bm = vq; bi = iq; }
        }
        smax[tid] = bm; sidx[tid] = bi;
    }
    __syncthreads();
    float rowmax = smax[r * 8];
    int   rowarg = sidx[r * 8];
    float lsum = 0.0f;
    for (int ci = j; ci < CHUNK; ci += 8)
        lsum += __expf(L[r * LPITCH + ci] - rowmax);
    __syncthreads();
    smax[tid] = lsum;
    __syncthreads();
    if (j == 0) {
        float s = 0.0f;
        for (int q = 0; q < 8; ++q) s += smax[tid + q];
        int o = row * NCH + ch;
        pmax[o] = rowmax;
        psum[o] = s;
        pidx[o] = rowarg;
        float tg = -1e30f;
        int tt = row % Tt, n = row / Tt;
        if (tt < Tt - 1) {
            int tok = target[n * Tt + tt + 1];
            if (tok >= colbase && tok < colbase + CHUNK)
                tg = L[r * LPITCH + (tok - colbase)];
        }
        ptgt[o] = tg;
    }
}

// ---------------- combine chunk partials per row -----------------------
__global__ __launch_bounds__(128) void finalize_kernel(const float* __restrict__ pmax,
                                                       const float* __restrict__ psum,
                                                       const int* __restrict__ pidx,
                                                       const float* __restrict__ ptgt,
                                                       const int* __restrict__ target,
                                                       float* __restrict__ rownll,
                                                       float* __restrict__ dout) {
    __shared__ float sv[128]; __shared__ int si[128];
    __shared__ float ss[128]; __shared__ float st[128];
    int row = blockIdx.x, tid = threadIdx.x;
    float v = -1e30f; int ix = 0x7fffffff; float tg = -1e30f;
    if (tid < NCH) {
        int o = row * NCH + tid;
        v = pmax[o]; ix = pidx[o]; tg = ptgt[o];
    }
    sv[tid] = v; si[tid] = ix; st[tid] = tg;
    __syncthreads();
    for (int off = 64; off > 0; off >>= 1) {
        if (tid < off) {
            float v2 = sv[tid + off]; int i2 = si[tid + off];
            if (v2 > sv[tid] || (v2 == sv[tid] && i2 < si[tid])) { sv[tid] = v2; si[tid] = i2; }
            st[tid] = fmaxf(st[tid], st[tid + off]);
        }
        __syncthreads();
    }
    float gmax = sv[0]; int gidx = si[0]; float tgtv = st[0];
    float s = 0.0f;
    if (tid < NCH) s = psum[row * NCH + tid] * __expf(pmax[row * NCH + tid] - gmax);
    ss[tid] = s;
    __syncthreads();
    for (int off = 64; off > 0; off >>= 1) {
        if (tid < off) ss[tid] += ss[tid + off];
        __syncthreads();
    }
    if (tid == 0) {
        float logZ = gmax + __logf(ss[0]);
        dout[1 + row] = (float)gidx;      // preds, row-major (n*T + t)
        int tt = row % Tt, n = row / Tt;
        float nll = 0.0f;
        if (tt < Tt - 1) {
            int tok = target[n * Tt + tt + 1];
            if (tok != 0) nll = logZ - tgtv;
        }
        rownll[row] = nll;
    }
}

__global__ __launch_bounds__(256) void loss_kernel(const float* __restrict__ rownll,
                                                   float* __restrict__ dout) {
    __shared__ float s[256];
    int tid = threadIdx.x;
    float a = 0.0f;
    for (int i = tid; i < ROWS; i += 256) a += rownll[i];
    s[tid] = a;
    __syncthreads();
    for (int off = 128; off > 0; off >>= 1) {
        if (tid < off) s[tid] += s[tid + off];
        __syncthreads();
    }
    if (tid == 0) dout[0] = s[0] / (float)Nn;
}

// ---------------- host orchestration -----------------------------------
extern "C" void kernel_launch(void* const* d_in, const int* in_sizes, int n_in,
                              void* d_out, int out_size, void* d_ws, size_t ws_size,
                              hipStream_t stream) {
    const float* enc     = (const float*)d_in[0];
    const float* h0      = (const float*)d_in[1];
    const float* c0      = (const float*)d_in[2];
    const int*   src_len = (const int*)  d_in[3];
    const int*   target  = (const int*)  d_in[4];
    const float* emb     = (const float*)d_in[5];
    const float* W_ih    = (const float*)d_in[6];
    const float* W_hh    = (const float*)d_in[7];
    const float* b_ih    = (const float*)d_in[8];
    const float* b_hh    = (const float*)d_in[9];
    const float* W_out   = (const float*)d_in[10];
    const float* b_out   = (const float*)d_in[11];
    float* dout = (float*)d_out;

    float* ws = (float*)d_ws;
    float* h       = ws;                 ws += Nn * Hh;
    float* c       = ws;                 ws += Nn * Hh;
    float* context = ws;                 ws += Nn * Hh;
    float* xbuf    = ws;                 ws += Nn * KX;
    float* gates   = ws;                 ws += Nn * Gg;
    float* dec_out = ws;                 ws += (size_t)ROWS * Hh;
    float* pmax    = ws;                 ws += (size_t)ROWS * NCH;
    float* psum    = ws;                 ws += (size_t)ROWS * NCH;
    float* ptgt    = ws;                 ws += (size_t)ROWS * NCH;
    int*   pidx    = (int*)ws;           ws += (size_t)ROWS * NCH;
    float* rownll  = ws;                 ws += ROWS;

    hipMemcpyAsync(h, h0, sizeof(float) * Nn * Hh, hipMemcpyDeviceToDevice, stream);
    hipMemcpyAsync(c, c0, sizeof(float) * Nn * Hh, hipMemcpyDeviceToDevice, stream);

    for (int t = 0; t < Tt; ++t) {
        attn_kernel<<<Nn, Ss, 0, stream>>>(enc, src_len, h, context);
        xpack_kernel<<<(Nn * KX + 255) / 256, 256, 0, stream>>>(emb, target, context, h, xbuf, t);
        gates_gemm_kernel<<<32, 256, 0, stream>>>(xbuf, W_ih, W_hh, gates);
        lstm_update_kernel<<<(Nn * Hh + 255) / 256, 256, 0, stream>>>(gates, b_ih, b_hh,
                                                                      h, c, dec_out, t);
    }
    logits_kernel<<<dim3(NCH, ROWS / ROWT), 256, 0, stream>>>(dec_out, W_out, b_out, target,
                                                              pmax, psum, pidx, ptgt);
    finalize_kernel<<<ROWS, 128, 0, stream>>>(pmax, psum, pidx, ptgt, target, rownll, dout);
    loss_kernel<<<1, 256, 0, stream>>>(rownll, dout);
}